// KattentionV4_87625922773651
// MI455X (gfx1250) — compile-verified
//
#include <hip/hip_runtime.h>
#include <math.h>

#define C_  4
#define K_  10
#define H_  32
#define L_  1024
#define LW  1015            // L - K + 1
#define D_  40              // K_*C_
#define B_  2

typedef __attribute__((ext_vector_type(2))) float v2f;
typedef __attribute__((ext_vector_type(8))) float v8f;

// ---------------------------------------------------------------------------
// Kernel 1: build Q_W and Kmat.
//   Kmat[b,l,:] = X[b, l*4 : l*4+40]   (contiguous slice, C=4)
//   Q_W[b,l,h,g*4+o] = sum_i X[b,l+g,i] * W[(g*H+h)*16 + o*4 + i]
// One thread per (b,l,h): 160 FMAs, float4-vectorized I/O.
// ---------------------------------------------------------------------------
__global__ __launch_bounds__(256)
void kattn_prep_kernel(const float* __restrict__ X, const float* __restrict__ W,
                       float* __restrict__ qw, float* __restrict__ kmat) {
    int t = blockIdx.x * blockDim.x + threadIdx.x;
    const int total = B_ * LW * H_;
    if (t >= total) return;
    int h  = t % H_;
    int bl = t / H_;
    int l  = bl % LW;
    int b  = bl / LW;

    const float* xp = X + ((size_t)b * L_ + l) * C_;   // 40 contiguous floats, 16B aligned
    float x[D_];
    #pragma unroll
    for (int j = 0; j < 10; ++j) {
        float4 v = ((const float4*)xp)[j];
        x[4*j+0] = v.x; x[4*j+1] = v.y; x[4*j+2] = v.z; x[4*j+3] = v.w;
    }

    float out[D_];
    #pragma unroll
    for (int g = 0; g < K_; ++g) {
        const float* wb = W + ((size_t)g * H_ + h) * 16;   // W[g,h,:,:] contiguous 16 floats
        #pragma unroll
        for (int o = 0; o < 4; ++o) {
            float acc = 0.f;
            #pragma unroll
            for (int i = 0; i < 4; ++i)
                acc = fmaf(x[g*4 + i], wb[o*4 + i], acc);
            out[g*4 + o] = acc;
        }
    }

    float* qp = qw + ((size_t)((size_t)b * LW + l) * H_ + h) * D_;
    #pragma unroll
    for (int j = 0; j < 10; ++j) {
        float4 v;
        v.x = out[4*j+0]; v.y = out[4*j+1]; v.z = out[4*j+2]; v.w = out[4*j+3];
        ((float4*)qp)[j] = v;
    }

    if (h == 0) {
        float* kp = kmat + ((size_t)b * LW + l) * D_;
        #pragma unroll
        for (int j = 0; j < 10; ++j)
            ((float4*)kp)[j] = ((const float4*)xp)[j];
    }
}

// ---------------------------------------------------------------------------
// Kernel 2: fused logits GEMM (f32 WMMA 16x16x4) + row softmax.
// One block per (b, h, 16-row q-tile). 8 waves; each wave owns column tiles
// ct = wave, wave+8, ... (64 tiles cover 1015 cols). Accumulated 16x16 tiles
// land in a 16x1024 LDS strip (64 KB); then 16 lanes per row do the softmax.
// ---------------------------------------------------------------------------
__global__ __launch_bounds__(256)
void kattn_logits_softmax_kernel(const float* __restrict__ qw,
                                 const float* __restrict__ kmat,
                                 float* __restrict__ probs,
                                 float* __restrict__ logits) {
    __shared__ float strip[16][1024];   // 64 KB

    const int qt   = blockIdx.x;        // 0..63
    const int h    = blockIdx.y;        // 0..31
    const int b    = blockIdx.z;        // 0..1
    const int tid  = threadIdx.x;
    const int lane = tid & 31;
    const int wave = tid >> 5;
    const int half = lane >> 4;         // selects K-pair (A/B frag) / M+8 (C frag)
    const int l16  = lane & 15;
    const int q0   = qt * 16;

    // ---- Preload the 10 A fragments for this wave's 16 q-rows -------------
    // A layout (16x4 f32): lane m in [0,16): VGPR0=A[m][2*half], VGPR1=A[m][2*half+1]
    {
        int qrow = q0 + l16;
        int qc   = qrow < LW ? qrow : LW - 1;        // clamp: keep EXEC all-ones
        const float* ap = qw + ((size_t)((size_t)b * LW + qc) * H_ + h) * D_ + 2 * half;

        v2f afrag[10];
        #pragma unroll
        for (int s = 0; s < 10; ++s)
            afrag[s] = *(const v2f*)(ap + 4 * s);

        const float* kb = kmat + (size_t)b * LW * D_;

        for (int ct = wave; ct < 64; ct += 8) {
            int col = ct * 16 + l16;
            int kc  = col < LW ? col : LW - 1;       // clamp column load
            const float* bp = kb + (size_t)kc * D_ + 2 * half;

            v8f acc = {};
            #pragma unroll
            for (int s = 0; s < 10; ++s) {
                v2f bfrag = *(const v2f*)(bp + 4 * s);
                acc = __builtin_amdgcn_wmma_f32_16x16x4_f32(
                        /*neg_a=*/false, afrag[s],
                        /*neg_b=*/false, bfrag,
                        /*c_mod=*/(short)0, acc,
                        /*reuse_a=*/false, /*reuse_b=*/false);
            }

            // C layout: VGPR r -> row M = r + 8*half, col N = lane&15
            #pragma unroll
            for (int r = 0; r < 8; ++r)
                strip[r + 8 * half][ct * 16 + l16] = acc[r];
        }
    }
    __syncthreads();

    // ---- Softmax: 16 threads per row, rows 0..15 --------------------------
    const int row = tid >> 4;
    const int sub = tid & 15;
    const int q   = q0 + row;
    float* lrow = &strip[row][0];
    const size_t gbase = (((size_t)b * H_ + h) * LW + (size_t)q) * (size_t)LW;
    const bool qvalid = (q < LW);

    // pass 1: row max + stream logits to global
    float m = -__builtin_inff();
    for (int c = sub; c < LW; c += 16) {
        float v = lrow[c];
        if (qvalid) logits[gbase + c] = v;
        m = fmaxf(m, v);
    }
    #pragma unroll
    for (int i = 1; i < 16; i <<= 1)
        m = fmaxf(m, __shfl_xor(m, i, 16));

    // pass 2: exp + row sum (store exp back into LDS)
    float s = 0.f;
    for (int c = sub; c < LW; c += 16) {
        float e = __expf(lrow[c] - m);
        lrow[c] = e;
        s += e;
    }
    #pragma unroll
    for (int i = 1; i < 16; i <<= 1)
        s += __shfl_xor(s, i, 16);
    float inv = 1.0f / s;

    // pass 3: normalized probs to global
    if (qvalid) {
        for (int c = sub; c < LW; c += 16)
            probs[gbase + c] = lrow[c] * inv;
    }
}

// ---------------------------------------------------------------------------
// d_out layout (reference return order, flat):
//   [0, S1)         attn_probs   S1 = B_*H_*LW*LW
//   [S1, 2*S1)      attn_logits
//   [2*S1, +QW)     Q_W          QW = B_*LW*H_*D_
//   [.., +KM)       Kmat         KM = B_*LW*D_
// ---------------------------------------------------------------------------
extern "C" void kernel_launch(void* const* d_in, const int* in_sizes, int n_in,
                              void* d_out, int out_size, void* d_ws, size_t ws_size,
                              hipStream_t stream) {
    (void)in_sizes; (void)n_in; (void)out_size; (void)d_ws; (void)ws_size;

    const float* X = (const float*)d_in[0];
    const float* W = (const float*)d_in[1];
    float* out = (float*)d_out;

    const size_t S1 = (size_t)B_ * H_ * LW * LW;
    float* probs  = out;
    float* logits = out + S1;
    float* qw     = out + 2 * S1;
    float* kmat   = qw + (size_t)B_ * LW * H_ * D_;

    const int total = B_ * LW * H_;
    kattn_prep_kernel<<<(total + 255) / 256, 256, 0, stream>>>(X, W, qw, kmat);

    dim3 grid(64, H_, B_);   // q-tiles x heads x batch
    kattn_logits_softmax_kernel<<<grid, 256, 0, stream>>>(qw, kmat, probs, logits);
}